// DecorelationNormalization_17059610100108
// MI455X (gfx1250) — compile-verified
//
#include <hip/hip_runtime.h>

typedef __attribute__((ext_vector_type(2))) float v2f;
typedef __attribute__((ext_vector_type(8))) float v8f;

#define CCH   256                     // channels
#define NROWS 131072                  // 32*64*64 samples
#define EPSV 0.001f

// Gram kernel blocking: 32x32 register blocks (2x2 WMMA tiles), upper block tri
#define GBLOCKS 36                    // 8x8 grid of 32x32 blocks, bi<=bj
#define GK_CHUNKS 128
#define GK_ROWS (NROWS / GK_CHUNKS)   // 1024 rows per chunk

// workspace layout (float offsets)
#define WS_MEAN 0
#define WS_G    256
#define WS_A2   (256 + 65536)
#define WS_W    (256 + 2 * 65536)
#define WS_WM   (256 + 3 * 65536)
#define WS_ZERO_N (256 + 65536)       // region to zero: mean sums + raw Gram

// ---------------------------------------------------------------- utilities
__global__ void dbn_zero(float* __restrict__ p, int n) {
    int i = blockIdx.x * blockDim.x + threadIdx.x;
    if (i < n) p[i] = 0.0f;
}

// column sums of X (n x 256): one block = 256 rows, thread t owns channel t
__global__ void dbn_colsum(const float* __restrict__ x, float* __restrict__ sums) {
    int t = threadIdx.x;
    int r0 = blockIdx.x * 256;
    float acc = 0.0f;
    const float* p = x + r0 * CCH + t;
    #pragma unroll 4
    for (int r = 0; r < 256; ++r) acc += p[r * CCH];
    atomicAdd(&sums[t], acc);
}

__global__ void dbn_meandiv(float* __restrict__ m) {
    m[threadIdx.x] *= (1.0f / (float)NROWS);
}

// ------------------------------------------------------- Gram = X^T X (WMMA)
// one wave computes a 32x32 block (2x2 WMMA tiles) over one K-chunk.
// 2x2 blocking shares A/B fragments: 8 dword loads per 4 WMMAs.
__global__ void dbn_gram(const float* __restrict__ x, float* __restrict__ G) {
    int wave = threadIdx.x >> 5;
    int lane = threadIdx.x & 31;
    int id = blockIdx.x * (blockDim.x >> 5) + wave;   // 0 .. GBLOCKS*GK_CHUNKS-1
    int blk = id % GBLOCKS;
    int chunk = id / GBLOCKS;

    // decode blk -> (bi, bj), bi <= bj over an 8x8 grid
    int bi = 0, rem = blk;
    while (rem >= 8 - bi) { rem -= 8 - bi; ++bi; }
    int bj = bi + rem;
    int ci = bi * 32, cj = bj * 32;

    int l15  = lane & 15;
    int koff = (lane < 16) ? 0 : 2;      // A/B K-wrap at lane 16
    int k0 = chunk * GK_ROWS;

    const float* pa0 = x + ci + l15;          // A tiles: M = channels ci..ci+31
    const float* pa1 = pa0 + 16;
    const float* pb0 = x + cj + l15;          // B tiles: N = channels cj..cj+31
    const float* pb1 = pb0 + 16;

    v8f c00 = {}, c01 = {}, c10 = {}, c11 = {};
    #pragma unroll 4
    for (int k = 0; k < GK_ROWS; k += 4) {
        int roff = (k0 + k + koff) * CCH;
        v2f a0, a1, b0, b1;
        a0.x = pa0[roff];  a0.y = pa0[roff + CCH];
        a1.x = pa1[roff];  a1.y = pa1[roff + CCH];
        b0.x = pb0[roff];  b0.y = pb0[roff + CCH];
        b1.x = pb1[roff];  b1.y = pb1[roff + CCH];
        c00 = __builtin_amdgcn_wmma_f32_16x16x4_f32(false, a0, false, b0, (short)0, c00, false, false);
        c01 = __builtin_amdgcn_wmma_f32_16x16x4_f32(false, a0, false, b1, (short)0, c01, false, false);
        c10 = __builtin_amdgcn_wmma_f32_16x16x4_f32(false, a1, false, b0, (short)0, c10, false, false);
        c11 = __builtin_amdgcn_wmma_f32_16x16x4_f32(false, a1, false, b1, (short)0, c11, false, false);
    }

    // D layout: VGPR r -> (M = r | r+8, N = lane&15)
    int mbase = (lane < 16) ? 0 : 8;
    #pragma unroll
    for (int r = 0; r < 8; ++r) {
        int r0 = ci + mbase + r;              // tile rows 0..15
        int r1 = ci + 16 + mbase + r;         // tile rows 16..31
        int cA = cj + l15;
        int cB = cj + 16 + l15;
        atomicAdd(&G[r0 * CCH + cA], c00[r]);
        atomicAdd(&G[r0 * CCH + cB], c01[r]);
        atomicAdd(&G[r1 * CCH + cA], c10[r]);
        atomicAdd(&G[r1 * CCH + cB], c11[r]);
    }
}

// mirror block-lower from block-upper, subtract mean outer product, shrink
__global__ void dbn_shrink(const float* __restrict__ G, const float* __restrict__ m,
                           float* __restrict__ A2) {
    int idx = blockIdx.x * blockDim.x + threadIdx.x;   // 0..65535
    int i = idx >> 8, j = idx & 255;
    int bi = i >> 5, bj = j >> 5;                      // 32x32 block coords
    float g = (bi > bj) ? G[j * CCH + i] : G[i * CCH + j];
    float cov = (g - (float)NROWS * m[i] * m[j]) * (1.0f / ((float)NROWS - 1.0f));
    A2[i * CCH + j] = (1.0f - EPSV) * cov + ((i == j) ? EPSV : 0.0f);
}

// ----------------------------------------------- Cholesky (in place, lower)
__global__ void dbn_cholesky(float* __restrict__ A) {
    int t = threadIdx.x;
    for (int k = 0; k < CCH; ++k) {
        if (t == 0) A[k * CCH + k] = sqrtf(A[k * CCH + k]);
        __syncthreads();
        float dk = A[k * CCH + k];
        int i = k + 1 + t;
        if (i < CCH) A[i * CCH + k] /= dk;
        __syncthreads();
        if (i < CCH) {
            float lik = A[i * CCH + k];
            for (int j = k + 1; j <= i; ++j)
                A[i * CCH + j] -= lik * A[j * CCH + k];
        }
        __syncthreads();
    }
}

// W = L^{-1} (row-major, full 256x256 with zero upper); thread j = column j
__global__ void dbn_trinv(const float* __restrict__ L, float* __restrict__ W) {
    int j = threadIdx.x;
    for (int i = 0; i < j; ++i) W[i * CCH + j] = 0.0f;
    W[j * CCH + j] = 1.0f / L[j * CCH + j];
    for (int i = j + 1; i < CCH; ++i) {
        float s = 0.0f;
        for (int k = j; k < i; ++k) s += L[i * CCH + k] * W[k * CCH + j];
        W[i * CCH + j] = -s / L[i * CCH + i];
    }
}

__global__ void dbn_wm(const float* __restrict__ W, const float* __restrict__ m,
                       float* __restrict__ WM) {
    int i = threadIdx.x;
    float s = 0.0f;
    for (int j = 0; j <= i; ++j) s += W[i * CCH + j] * m[j];
    WM[i] = s;
}

// ------------------------------------- whitening: out = X W^T - (W m) (WMMA)
// one wave -> 16 rows x 64 output channels (4 accumulators share A fragment)
__global__ void dbn_whiten(const float* __restrict__ x, const float* __restrict__ W,
                           const float* __restrict__ WM, float* __restrict__ out) {
    int wave = threadIdx.x >> 5;
    int lane = threadIdx.x & 31;
    int id = blockIdx.x * (blockDim.x >> 5) + wave;   // 0 .. 32767
    int cg    = id & 3;          // 64-channel output group
    int rtile = id >> 2;         // 0 .. 8191 (16-row tile)

    int l15  = lane & 15;
    int koff = (lane < 16) ? 0 : 2;
    int mbase = (lane < 16) ? 0 : 8;

    const float* pa  = x + (rtile * 16 + l15) * CCH + koff;
    const float* pb0 = W + (cg * 64 +  0 + l15) * CCH + koff;
    const float* pb1 = W + (cg * 64 + 16 + l15) * CCH + koff;
    const float* pb2 = W + (cg * 64 + 32 + l15) * CCH + koff;
    const float* pb3 = W + (cg * 64 + 48 + l15) * CCH + koff;

    v8f c0 = {}, c1 = {}, c2 = {}, c3 = {};
    #pragma unroll 4
    for (int j = 0; j < CCH; j += 4) {
        v2f a  = *(const v2f*)(pa  + j);
        v2f b0 = *(const v2f*)(pb0 + j);
        v2f b1 = *(const v2f*)(pb1 + j);
        v2f b2 = *(const v2f*)(pb2 + j);
        v2f b3 = *(const v2f*)(pb3 + j);
        c0 = __builtin_amdgcn_wmma_f32_16x16x4_f32(false, a, false, b0, (short)0, c0, false, false);
        c1 = __builtin_amdgcn_wmma_f32_16x16x4_f32(false, a, false, b1, (short)0, c1, false, false);
        c2 = __builtin_amdgcn_wmma_f32_16x16x4_f32(false, a, false, b2, (short)0, c2, false, false);
        c3 = __builtin_amdgcn_wmma_f32_16x16x4_f32(false, a, false, b3, (short)0, c3, false, false);
    }

    float wm0 = WM[cg * 64 +  0 + l15];
    float wm1 = WM[cg * 64 + 16 + l15];
    float wm2 = WM[cg * 64 + 32 + l15];
    float wm3 = WM[cg * 64 + 48 + l15];

    #pragma unroll
    for (int r = 0; r < 8; ++r) {
        int row = rtile * 16 + mbase + r;
        float* po = out + row * CCH + cg * 64 + l15;
        po[ 0] = c0[r] - wm0;
        po[16] = c1[r] - wm1;
        po[32] = c2[r] - wm2;
        po[48] = c3[r] - wm3;
    }
}

// ---------------------------------------------------------------- launcher
extern "C" void kernel_launch(void* const* d_in, const int* in_sizes, int n_in,
                              void* d_out, int out_size, void* d_ws, size_t ws_size,
                              hipStream_t stream) {
    (void)in_sizes; (void)n_in; (void)out_size; (void)ws_size;
    const float* x = (const float*)d_in[0];
    float* out = (float*)d_out;
    float* ws  = (float*)d_ws;

    float* mean = ws + WS_MEAN;
    float* G    = ws + WS_G;
    float* A2   = ws + WS_A2;
    float* W    = ws + WS_W;
    float* WM   = ws + WS_WM;

    dbn_zero<<<(WS_ZERO_N + 255) / 256, 256, 0, stream>>>(ws, WS_ZERO_N);
    dbn_colsum<<<NROWS / 256, 256, 0, stream>>>(x, mean);
    dbn_meandiv<<<1, 256, 0, stream>>>(mean);
    dbn_gram<<<GBLOCKS * GK_CHUNKS / 8, 256, 0, stream>>>(x, G);
    dbn_shrink<<<256, 256, 0, stream>>>(G, mean, A2);
    dbn_cholesky<<<1, 256, 0, stream>>>(A2);
    dbn_trinv<<<1, 256, 0, stream>>>(A2, W);
    dbn_wm<<<1, 256, 0, stream>>>(W, mean, WM);
    dbn_whiten<<<NROWS / 16 * 4 / 8, 256, 0, stream>>>(x, W, WM, out);
}